// CovidModel_13262859010430
// MI455X (gfx1250) — compile-verified
//
#include <hip/hip_runtime.h>
#include <stdint.h>

#define J 10
#define S_SAMPLES 1000
#define T_DAYS 50000
#define CHUNK 256
#define NCHUNK ((T_DAYS + CHUNK - 1) / CHUNK)   /* 196 */
#define TILE_D 200                               /* divisible by J; T_DAYS/TILE_D tiles */
#define NTILE (T_DAYS / TILE_D)                  /* 250 */

typedef float v2f __attribute__((ext_vector_type(2)));
typedef float v8f __attribute__((ext_vector_type(8)));

// ---------------------------------------------------------------------------
// Kernel 1: per-chunk (256-elem) inclusive prefix sum of log2(r) via WMMA.
// Scan of a 16x16 column-major block == Ltri(16x16) x X(16x16), built from
// four chained v_wmma_f32_16x16x4_f32 (K accumulated through C).
// One wave per chunk; EXEC all ones as WMMA requires.
// Loads are branch-free: index clamped to T-1, pad value selected to 0
// (= log2(1)) afterwards, so no exec-mask branches around the loads.
// ---------------------------------------------------------------------------
__global__ __launch_bounds__(32) void scan_chunks(const float* __restrict__ r,
                                                  float* __restrict__ L2,
                                                  float* __restrict__ chunkSums) {
  const int lane = threadIdx.x;      // 0..31
  const int half = lane >> 4;        // 0 or 1
  const int col  = lane & 15;        // matrix column / row index
  const int base = blockIdx.x * CHUNK;

  v8f acc = {};
#pragma unroll
  for (int kk = 0; kk < 4; ++kk) {
    const int k0 = 4 * kk + 2 * half;
    // B operand (4x16 slice of X): v0 = X[k0][col], v1 = X[k0+1][col],
    // where X[k][c] = log2(r[base + c*16 + k]); padded region contributes 0.
    const int i0 = base + col * 16 + k0;
    const int c0 = (i0     < T_DAYS) ? i0     : (T_DAYS - 1);
    const int c1 = (i0 + 1 < T_DAYS) ? i0 + 1 : (T_DAYS - 1);
    const float x0 = __builtin_amdgcn_logf(r[c0]);   // v_log_f32 = log2
    const float x1 = __builtin_amdgcn_logf(r[c1]);
    v2f b;
    b.x = (i0     < T_DAYS) ? x0 : 0.0f;
    b.y = (i0 + 1 < T_DAYS) ? x1 : 0.0f;
    // A operand (16x4 slice of lower-triangular ones): A[m][k] = (k <= m)
    v2f a;
    a.x = (k0     <= col) ? 1.0f : 0.0f;
    a.y = (k0 + 1 <= col) ? 1.0f : 0.0f;
    acc = __builtin_amdgcn_wmma_f32_16x16x4_f32(false, a, false, b,
                                                (short)0, acc, false, false);
  }

  // Column totals = row M=15 of D, which lives in acc[7] of lanes 16..31.
  const float colTot = acc[7];
  float off = 0.0f, total = 0.0f;
#pragma unroll
  for (int c = 0; c < 16; ++c) {
    const float t = __shfl(colTot, 16 + c, 32);
    total += t;
    if (c < col) off += t;   // exclusive prefix of column totals
  }

  // D layout: lane(col)<16 holds rows 0..7 of its column; lane+16 rows 8..15.
#pragma unroll
  for (int v = 0; v < 8; ++v) {
    const int idx = base + col * 16 + half * 8 + v;
    if (idx < T_DAYS) L2[idx] = acc[v] + off;
  }
  if (lane == 0) chunkSums[blockIdx.x] = total;
}

// ---------------------------------------------------------------------------
// Kernel 2: serial f64 exclusive scan of the 196 chunk totals (trivial work,
// done in double to keep the 50000-long accumulation tight).
// ---------------------------------------------------------------------------
__global__ void scan_carries(const float* __restrict__ chunkSums,
                             float* __restrict__ carry) {
  if (threadIdx.x == 0 && blockIdx.x == 0) {
    double run = 0.0;
    for (int i = 0; i < NCHUNK; ++i) {
      carry[i] = (float)run;
      run += (double)chunkSums[i];
    }
  }
}

// Kernel 3: fold carries back so L2[d] = full inclusive cumsum of log2(r).
__global__ __launch_bounds__(256) void add_carries(float* __restrict__ L2,
                                                   const float* __restrict__ carry) {
  const int i = blockIdx.x * blockDim.x + threadIdx.x;
  if (i < T_DAYS) L2[i] += carry[i >> 8];
}

// ---------------------------------------------------------------------------
// Kernel 4: main forecast. threads span samples (coalesced stores), blocks own
// 200-day tiles. Tile's L2 window staged to LDS with the CDNA5 async
// global->LDS path. Rolling J-register window; day loop unrolled by J so all
// circular indices are compile-time constants.
// ---------------------------------------------------------------------------
__global__ __launch_bounds__(256) void forecast(const float* __restrict__ L2,
                                                const float* __restrict__ warmup,
                                                const float* __restrict__ Tser,
                                                const float* __restrict__ rho,
                                                const float* __restrict__ pi,
                                                float* __restrict__ out) {
  __shared__ float Lsh[TILE_D + J];
  const int tid = threadIdx.x;
  const int s   = blockIdx.x * blockDim.x + tid;
  const int d0  = blockIdx.y * TILE_D;

  // Async global->LDS copy of L2[d0-J .. d0+TILE_D) (uniform across block).
  {
    const int src = d0 - J + tid;
    if (tid < (TILE_D + J) && src >= 0) {
      const unsigned lds_off = (unsigned)(uintptr_t)&Lsh[tid];  // low 32b = LDS byte addr
      const float* gp = L2 + src;
      asm volatile("global_load_async_to_lds_b32 %0, %1, off"
                   :: "v"(lds_off), "v"(gp) : "memory");
    }
    asm volatile("s_wait_asynccnt 0" ::: "memory");
  }
  __syncthreads();

  if (s >= S_SAMPLES) return;   // no further barriers below

  const float A0   = warmup[(J - 1) * S_SAMPLES + s];
  const float invT = 1.0f / Tser[s];
  const float rh   = rho[s];

  float w[J];
#pragma unroll
  for (int j = 0; j < J; ++j) w[j] = rh * pi[j * S_SAMPLES + s];

  // Window: W[j] = A_ext value for day (d0 - 1 - j)
  float W[J];
  if (d0 == 0) {
#pragma unroll
    for (int j = 0; j < J; ++j) W[j] = warmup[(J - 1 - j) * S_SAMPLES + s];
  } else {
#pragma unroll
    for (int j = 0; j < J; ++j)
      W[j] = A0 * __builtin_amdgcn_exp2f(Lsh[J - 1 - j] * invT);  // v_exp_f32 = exp2
  }

  float* o = out + (size_t)d0 * S_SAMPLES + s;
  for (int t = 0; t < TILE_D; t += J) {
#pragma unroll
    for (int u = 0; u < J; ++u) {
      // M[d] uses window BEFORE inserting day d's A value.
      float m = 0.0f;
#pragma unroll
      for (int j = 0; j < J; ++j) m += w[j] * W[(j - u + J) % J];
      const float a = A0 * __builtin_amdgcn_exp2f(Lsh[t + u + J] * invT);
      W[J - 1 - u] = a;                       // replace oldest (static slot)
      o[(size_t)(t + u) * S_SAMPLES] = m;
    }
  }
}

extern "C" void kernel_launch(void* const* d_in, const int* in_sizes, int n_in,
                              void* d_out, int out_size, void* d_ws, size_t ws_size,
                              hipStream_t stream) {
  (void)in_sizes; (void)n_in; (void)out_size; (void)ws_size;
  const float* r_t  = (const float*)d_in[0];   // (1, T)
  const float* warm = (const float*)d_in[1];   // (J, S)
  const float* Tser = (const float*)d_in[2];   // (S,)
  const float* rho  = (const float*)d_in[3];   // (S,)
  const float* pi   = (const float*)d_in[4];   // (J, S)
  float* out = (float*)d_out;                  // (T, S)

  float* L2        = (float*)d_ws;             // T floats
  float* chunkSums = L2 + T_DAYS;              // NCHUNK floats
  float* carry     = chunkSums + NCHUNK;       // NCHUNK floats

  scan_chunks<<<NCHUNK, 32, 0, stream>>>(r_t, L2, chunkSums);
  scan_carries<<<1, 32, 0, stream>>>(chunkSums, carry);
  add_carries<<<(T_DAYS + 255) / 256, 256, 0, stream>>>(L2, carry);

  dim3 grid((S_SAMPLES + 255) / 256, NTILE);
  forecast<<<grid, 256, 0, stream>>>(L2, warm, Tser, rho, pi, out);
}